// gate_36146444763231
// MI455X (gfx1250) — compile-verified
//
#include <hip/hip_runtime.h>

// out[o,b,v] = sum_j sum_u x[j,b,u] * W[o,j,u,v] * z[o,j,v]
// z = clip(sigmoid(log(u/(1-u)) + log(alpha)/BETA)*(EPS-GAMMA)+GAMMA, 0, 1)
//
// Compute-bound f32 problem (AI ~290 FLOP/B) -> V_WMMA_F32_16X16X4_F32 path.
// W slab is staged in LDS k-pair interleaved ([k/2][n*2+(k&1)]) so both A and B
// WMMA fragments are single contiguous ds_load_b64s (no regrouping v_movs).

typedef __attribute__((ext_vector_type(2))) float v2f;
typedef __attribute__((ext_vector_type(4))) float v4f;
typedef __attribute__((ext_vector_type(8))) float v8f;

#define I_DIM 8
#define O_DIM 4
#define U_DIM 256
#define B_DIM 4096

#define BM 64       // workgroup M tile (batch)
#define BN 256      // workgroup N tile = full U
#define KS 32       // K slab per staging round
#define LDX 36      // x slab row stride in dwords (16B-aligned rows, bank-staggered)
#define LDWP 544    // W pair-slab row stride in dwords (2*256 + 32: half-waves on disjoint banks)

__global__ __launch_bounds__(256)
void moe_gate_gemm_f32wmma(const float* __restrict__ xg,
                           const float* __restrict__ ugate,
                           const float* __restrict__ Wg,
                           const float* __restrict__ alpha,
                           float* __restrict__ outg)
{
    __shared__ float ldsX[BM * LDX];          // 9216 B : x[j] slab, [m][k]
    __shared__ float ldsWp[(KS / 2) * LDWP];  // 34816 B: z-scaled W slab, k-pair interleaved
    __shared__ float zbuf[BN];                // 1024 B : gate for this (o,j)

    const int tid  = threadIdx.x;
    const int o    = blockIdx.y;
    const int m0   = blockIdx.x * BM;

    const int wave = tid >> 5;
    const int lane = tid & 31;
    const int lh   = lane >> 4;        // half-wave: K+2 (A/B), M+8 (C/D) per ISA layout
    const int lm   = lane & 15;
    const int wm   = (wave >> 2) * 32; // wave M offset: 0 / 32
    const int wn   = (wave & 3) * 64;  // wave N offset: 0..192

    const float la = __logf(alpha[0]) * (1.0f / 0.9f);   // log(alpha)/BETA

    v8f acc[2][4] = {};                // 32x64 per wave: eight 16x16 f32 accumulators

    for (int j = 0; j < I_DIM; ++j) {
        // ---- gate z for this (o, j), all 256 columns ----
        {
            float uv = ugate[(o * I_DIM + j) * U_DIM + tid];
            float t  = __logf(uv) - __logf(1.0f - uv) + la;
            float s  = 1.0f / (1.0f + __expf(-t));
            float z  = s * 1.2f - 0.1f;                  // *(EPS-GAMMA)+GAMMA
            zbuf[tid] = fminf(fmaxf(z, 0.0f), 1.0f);
        }
        __syncthreads();

        const float* xj = xg + (size_t)j * B_DIM * U_DIM + (size_t)m0 * U_DIM;
        const float* Wj = Wg + ((size_t)(o * I_DIM + j) * U_DIM) * U_DIM;

        for (int ks = 0; ks < U_DIM; ks += KS) {
            // ---- stage x slab: 64x32 f32, b128 global loads -> b128 ds stores ----
            #pragma unroll
            for (int s = 0; s < 2; ++s) {
                int idx = tid + s * 256;              // 0..511 float4s
                int row = idx >> 3;
                int kq  = (idx & 7) << 2;
                v4f v = *(const v4f*)(xj + (size_t)row * U_DIM + ks + kq);
                *(v4f*)(&ldsX[row * LDX + kq]) = v;
            }
            // ---- stage W slab: 32x256 f32, gate folded in, k-pair interleaved ----
            // cell (kp, n): { W[2kp][n]z, W[2kp+1][n]z, W[2kp][n+1]z', W[2kp+1][n+1]z' }
            #pragma unroll
            for (int s = 0; s < 8; ++s) {
                int idx = tid + s * 256;              // 0..2047 cells
                int kp  = idx >> 7;                   // 0..15 k-pair row
                int n   = (idx & 127) << 1;           // 0..254 even column
                const float* wrow = Wj + (size_t)(ks + 2 * kp) * U_DIM + n;
                v2f g0 = *(const v2f*)(wrow);          // row k,   cols n..n+1
                v2f g1 = *(const v2f*)(wrow + U_DIM);  // row k+1, cols n..n+1
                float z0 = zbuf[n], z1 = zbuf[n + 1];
                v4f cell;
                cell.x = g0.x * z0; cell.y = g1.x * z0;
                cell.z = g0.y * z1; cell.w = g1.y * z1;
                *(v4f*)(&ldsWp[kp * LDWP + n * 2]) = cell;
            }
            // prefetch next K slab into cache while WMMAs run
            if (ks + KS < U_DIM) {
                __builtin_prefetch(xj + (size_t)(tid >> 2) * U_DIM + ks + KS + ((tid & 3) << 3), 0, 3);
                __builtin_prefetch(Wj + (size_t)(ks + KS + (tid >> 3)) * U_DIM + ((tid & 7) << 5), 0, 3);
            }
            __syncthreads();

            // ---- 16x16x4 f32 WMMA over the slab: 8 WMMAs per 6 b64 fragment loads ----
            #pragma unroll
            for (int k = 0; k < KS; k += 4) {
                // A 16x4: lane=M row, {K,K+1} contiguous (half-wave reads K+2,K+3)
                v2f a0 = *(const v2f*)(&ldsX[(wm + 0  + lm) * LDX + k + 2 * lh]);
                v2f a1 = *(const v2f*)(&ldsX[(wm + 16 + lm) * LDX + k + 2 * lh]);
                // B 4x16: lane=N col, {row K, row K+1} contiguous in pair layout
                const float* wb = &ldsWp[((k >> 1) + lh) * LDWP + (wn + lm) * 2];
                v2f b0 = *(const v2f*)(wb);
                v2f b1 = *(const v2f*)(wb + 32);
                v2f b2 = *(const v2f*)(wb + 64);
                v2f b3 = *(const v2f*)(wb + 96);

                acc[0][0] = __builtin_amdgcn_wmma_f32_16x16x4_f32(
                    false, a0, false, b0, (short)0, acc[0][0], false, false);
                acc[0][1] = __builtin_amdgcn_wmma_f32_16x16x4_f32(
                    false, a0, false, b1, (short)0, acc[0][1], false, false);
                acc[0][2] = __builtin_amdgcn_wmma_f32_16x16x4_f32(
                    false, a0, false, b2, (short)0, acc[0][2], false, false);
                acc[0][3] = __builtin_amdgcn_wmma_f32_16x16x4_f32(
                    false, a0, false, b3, (short)0, acc[0][3], false, false);
                acc[1][0] = __builtin_amdgcn_wmma_f32_16x16x4_f32(
                    false, a1, false, b0, (short)0, acc[1][0], false, false);
                acc[1][1] = __builtin_amdgcn_wmma_f32_16x16x4_f32(
                    false, a1, false, b1, (short)0, acc[1][1], false, false);
                acc[1][2] = __builtin_amdgcn_wmma_f32_16x16x4_f32(
                    false, a1, false, b2, (short)0, acc[1][2], false, false);
                acc[1][3] = __builtin_amdgcn_wmma_f32_16x16x4_f32(
                    false, a1, false, b3, (short)0, acc[1][3], false, false);
            }
            __syncthreads();
        }
    }

    // ---- store: C/D layout VGPR r -> M = r + 8*lh, N = lane%16 ----
    #pragma unroll
    for (int mi = 0; mi < 2; ++mi) {
        #pragma unroll
        for (int ni = 0; ni < 4; ++ni) {
            #pragma unroll
            for (int r = 0; r < 8; ++r) {
                int row = m0 + wm + mi * 16 + r + 8 * lh;
                int col = wn + ni * 16 + lm;
                outg[((size_t)o * B_DIM + row) * U_DIM + col] = acc[mi][ni][r];
            }
        }
    }
}

extern "C" void kernel_launch(void* const* d_in, const int* in_sizes, int n_in,
                              void* d_out, int out_size, void* d_ws, size_t ws_size,
                              hipStream_t stream) {
    const float* x     = (const float*)d_in[0];   // [I,B,U]
    const float* ugate = (const float*)d_in[1];   // [O,I,U]
    const float* W     = (const float*)d_in[2];   // [O,I,U,U]
    const float* alpha = (const float*)d_in[3];   // [1]
    float* out = (float*)d_out;                   // [O,B,U]

    dim3 grid(B_DIM / BM, O_DIM);                 // (64, 4)
    moe_gate_gemm_f32wmma<<<grid, 256, 0, stream>>>(x, ugate, W, alpha, out);
}